// VectorQuantizer_18700287606891
// MI455X (gfx1250) — compile-verified
//
#include <hip/hip_runtime.h>

typedef float v2f __attribute__((ext_vector_type(2)));
typedef float v8f __attribute__((ext_vector_type(8)));

#define VQ_N 65536
#define VQ_D 256
#define VQ_K 2048
#define VQ_BETA 0.25f
#define LDS_STRIDE 260  // 256 + 4 pad: bank-conflict-free float2 reads, 16B-aligned float4 stores

// ---------------------------------------------------------------------------
// Kernel 1: enorm[k] = sum_d emb[k][d]^2.  One wave per codebook row.
// ---------------------------------------------------------------------------
__global__ void __launch_bounds__(128) vq_enorm_kernel(const float* __restrict__ emb,
                                                       float* __restrict__ enorm) {
  const int lane = threadIdx.x & 31;
  const int wave = threadIdx.x >> 5;
  const int row  = blockIdx.x * 4 + wave;
  const float* e = emb + (size_t)row * VQ_D;
  float s = 0.f;
#pragma unroll
  for (int j = 0; j < VQ_D; j += 32) {
    float v = e[lane + j];
    s += v * v;
  }
#pragma unroll
  for (int m = 16; m >= 1; m >>= 1) s += __shfl_xor(s, m, 32);
  if (lane == 0) enorm[row] = s;
}

// ---------------------------------------------------------------------------
// Kernel 2: fused distance GEMM + argmin via V_WMMA_F32_16X16X4_F32.
// Block = 128 threads (4 waves). Each wave owns a 16-row slice of z (in LDS)
// and sweeps all 128 column tiles of emb, keeping a running per-row argmin of
// enorm[k] - 2 * (z . e_k).  (||z||^2 is row-constant -> dropped.)
// ---------------------------------------------------------------------------
__global__ void __launch_bounds__(128) vq_argmin_kernel(const float* __restrict__ z,
                                                        const float* __restrict__ emb,
                                                        const float* __restrict__ enorm,
                                                        int* __restrict__ idxOut) {
  __shared__ float lz[64 * LDS_STRIDE];
  const int tid  = threadIdx.x;
  const int lane = tid & 31;
  const int wave = tid >> 5;
  const int n16  = lane & 15;   // column within tile / A-row within slice
  const int half = lane >> 4;   // K-pair selector (A/B), row+8 selector (C)
  const int rowBlock = blockIdx.x * 64;

  // Cooperative stage: 64 x 256 z tile -> LDS (float4, 16B aligned).
  {
    const float4* zsrc = (const float4*)(z + (size_t)rowBlock * VQ_D);
    for (int i = tid; i < 64 * 64; i += 128) {
      const int r = i >> 6, c = i & 63;
      *(float4*)&lz[r * LDS_STRIDE + c * 4] = zsrc[r * 64 + c];
    }
  }
  __syncthreads();

  float bestVal[8];
  int   bestIdx[8];
#pragma unroll
  for (int r = 0; r < 8; ++r) { bestVal[r] = 3.4e38f; bestIdx[r] = 0; }

  // A operand source: this lane always reads z-row (wave*16 + n16), K-pair "half".
  const float* lzrow = &lz[(wave * 16 + n16) * LDS_STRIDE + 2 * half];

  for (int ct = 0; ct < VQ_K / 16; ++ct) {
    const int col = ct * 16 + n16;
    const float* eb = emb + (size_t)col * VQ_D + 2 * half;
    if (ct + 1 < VQ_K / 16)  // pull next emb column tile toward the WGP
      __builtin_prefetch(emb + (size_t)(col + 16) * VQ_D + lane * 8, 0, 0);

    v8f acc = {0.f, 0.f, 0.f, 0.f, 0.f, 0.f, 0.f, 0.f};
#pragma unroll 8
    for (int k = 0; k < VQ_D; k += 4) {
      v2f a = *(const v2f*)&lzrow[k];  // z[row][k + 2*half + {0,1}]
      v2f b = *(const v2f*)&eb[k];     // emb[col][k + 2*half + {0,1}] == B[k..][n]
      acc = __builtin_amdgcn_wmma_f32_16x16x4_f32(
          /*neg_a=*/false, a, /*neg_b=*/false, b,
          /*c_mod=*/(short)0, acc, /*reuse_a=*/false, /*reuse_b=*/false);
    }

    const float en = enorm[col];
#pragma unroll
    for (int r = 0; r < 8; ++r) {
      const float dv = en - 2.f * acc[r];  // C[r] = row (r + 8*half), col n16
      if (dv < bestVal[r]) { bestVal[r] = dv; bestIdx[r] = col; }
    }
  }

  // Cross-lane argmin over the 16 columns held in this half-wave.
#pragma unroll
  for (int r = 0; r < 8; ++r) {
    float v  = bestVal[r];
    int   bi = bestIdx[r];
#pragma unroll
    for (int m = 8; m >= 1; m >>= 1) {
      const float ov = __shfl_xor(v, m, 16);
      const int   oi = __shfl_xor(bi, m, 16);
      if (ov < v || (ov == v && oi < bi)) { v = ov; bi = oi; }  // first-min tie-break
    }
    if (n16 == 0) idxOut[rowBlock + wave * 16 + r + 8 * half] = bi;
  }
}

// ---------------------------------------------------------------------------
// Kernel 3: z_q = emb[idx] (this *is* z_q_st by straight-through identity),
// plus deterministic per-block partial sums of (z_q - z)^2.
// Block = 256 threads -> 4 rows, one float4 per thread.
// ---------------------------------------------------------------------------
__global__ void __launch_bounds__(256) vq_gather_kernel(const float* __restrict__ z,
                                                        const float* __restrict__ emb,
                                                        const int* __restrict__ idx,
                                                        float* __restrict__ out,
                                                        float* __restrict__ partial) {
  __shared__ float red[8];
  const int tid = threadIdx.x;
  const int row = blockIdx.x * 4 + (tid >> 6);
  const int c4  = tid & 63;
  const int k   = idx[row];
  const float4 e  = ((const float4*)(emb + (size_t)k * VQ_D))[c4];
  const float4 zv = ((const float4*)(z + (size_t)row * VQ_D))[c4];
  ((float4*)(out + (size_t)row * VQ_D))[c4] = e;
  const float dx = e.x - zv.x, dy = e.y - zv.y, dz = e.z - zv.z, dw = e.w - zv.w;
  float s = dx * dx + dy * dy + dz * dz + dw * dw;
#pragma unroll
  for (int m = 16; m >= 1; m >>= 1) s += __shfl_xor(s, m, 32);
  if ((tid & 31) == 0) red[tid >> 5] = s;
  __syncthreads();
  if (tid == 0) {
    float t = 0.f;
#pragma unroll
    for (int w = 0; w < 8; ++w) t += red[w];
    partial[blockIdx.x] = t;
  }
}

// ---------------------------------------------------------------------------
// Kernel 4: deterministic final reduction -> loss = (1+beta)*mean((z_q-z)^2).
// ---------------------------------------------------------------------------
__global__ void __launch_bounds__(256) vq_finalize_kernel(const float* __restrict__ partial,
                                                          float* __restrict__ lossOut) {
  __shared__ float red[8];
  const int tid = threadIdx.x;
  float s = 0.f;
  for (int i = tid; i < VQ_N / 4; i += 256) s += partial[i];
#pragma unroll
  for (int m = 16; m >= 1; m >>= 1) s += __shfl_xor(s, m, 32);
  if ((tid & 31) == 0) red[tid >> 5] = s;
  __syncthreads();
  if (tid == 0) {
    float t = 0.f;
#pragma unroll
    for (int w = 0; w < 8; ++w) t += red[w];
    lossOut[0] = (1.0f + VQ_BETA) * t / (float)((size_t)VQ_N * VQ_D);
  }
}

// ---------------------------------------------------------------------------
extern "C" void kernel_launch(void* const* d_in, const int* in_sizes, int n_in,
                              void* d_out, int out_size, void* d_ws, size_t ws_size,
                              hipStream_t stream) {
  (void)in_sizes; (void)n_in; (void)out_size; (void)ws_size;
  const float* z   = (const float*)d_in[0];  // [N, D]
  const float* emb = (const float*)d_in[1];  // [K, D]
  float* out = (float*)d_out;                // [N*D] z_q_st, then [1] loss

  float* ws      = (float*)d_ws;
  float* enorm   = ws;                        // [K]
  int*   idx     = (int*)(ws + VQ_K);         // [N]
  float* partial = ws + VQ_K + VQ_N;          // [N/4]

  vq_enorm_kernel  <<<VQ_K / 4, 128, 0, stream>>>(emb, enorm);
  vq_argmin_kernel <<<VQ_N / 64, 128, 0, stream>>>(z, emb, enorm, idx);
  vq_gather_kernel <<<VQ_N / 4, 256, 0, stream>>>(z, emb, idx, out, partial);
  vq_finalize_kernel<<<1, 256, 0, stream>>>(partial, out + (size_t)VQ_N * VQ_D);
}